// Eagle3OneModelWorker_16243566313554
// MI455X (gfx1250) — compile-verified
//
#include <hip/hip_runtime.h>
#include <hip/hip_bf16.h>

typedef __attribute__((ext_vector_type(2))) float v2f;
typedef __attribute__((ext_vector_type(8))) float v8f;

#define NCg   32
#define NGg   32
#define Lg    4
#define BSZ   64            // NC + NG
#define Hh    2048
#define VTg   50257
#define VDg   32000
#define NLROW 192           // NC + NG*(L+1)
#define NCTX  16384         // NC * CTX_LEN

// d_out layout (floats, concatenated in return order)
#define OUT_ACCEPTED  (NLROW * VTg)                 // 9,649,344
#define OUT_NUMACC    (OUT_ACCEPTED + BSZ*(Lg+1))
#define OUT_NEXTDRAFT (OUT_NUMACC + BSZ)
#define OUT_NEXTNEW   (OUT_NEXTDRAFT + BSZ*Lg)

__device__ __forceinline__ unsigned int f32_ord(float f) {
  unsigned int u = __float_as_uint(f);
  return (u & 0x80000000u) ? ~u : (u | 0x80000000u);
}

// ---------------------------------------------------------------------------
// Row-wise argmax over the big logits (192 x 50257), first-occurrence ties.
// ---------------------------------------------------------------------------
__global__ void argmax_rows(const float* __restrict__ logits, int ncols,
                            int* __restrict__ out_idx) {
  __shared__ float sv[256];
  __shared__ int   si[256];
  const int row = blockIdx.x;
  const float* p = logits + (size_t)row * ncols;
  float bv = -__builtin_inff();
  int bi = 0x7FFFFFFF;
  for (int c = threadIdx.x; c < ncols; c += blockDim.x) {
    float v = p[c];
    if (v > bv || (v == bv && c < bi)) { bv = v; bi = c; }
  }
  sv[threadIdx.x] = bv; si[threadIdx.x] = bi;
  __syncthreads();
  for (int s = 128; s > 0; s >>= 1) {
    if ((int)threadIdx.x < s) {
      float ov = sv[threadIdx.x + s]; int oc = si[threadIdx.x + s];
      if (ov > sv[threadIdx.x] || (ov == sv[threadIdx.x] && oc < si[threadIdx.x])) {
        sv[threadIdx.x] = ov; si[threadIdx.x] = oc;
      }
    }
    __syncthreads();
  }
  if (threadIdx.x == 0) out_idx[row] = si[0];
}

// ---------------------------------------------------------------------------
// Control logic: accepted, num_accepted, gather rows, initial ids, last_acc.
// ---------------------------------------------------------------------------
__global__ void control_kernel(const int* __restrict__ tt,
                               const int* __restrict__ draft_tokens,
                               const int* __restrict__ gather_ids_ctx,
                               float* __restrict__ out,
                               int* __restrict__ row_g,
                               int* __restrict__ ids,
                               unsigned long long* __restrict__ keys) {
  const int b = threadIdx.x;
  if (b >= BSZ) return;
  keys[b] = 0ull;
  int acc[Lg + 1];
  int na;
  if (b < NCg) {
    acc[0] = tt[b];
    #pragma unroll
    for (int j = 1; j <= Lg; ++j) acc[j] = 0;
    na = 1;
  } else {
    const int g = b - NCg;
    #pragma unroll
    for (int j = 0; j <= Lg; ++j) acc[j] = tt[NCg + g * (Lg + 1) + j];
    int ok = 1, m = 0;
    #pragma unroll
    for (int i = 0; i < Lg; ++i) {
      ok &= (draft_tokens[g * Lg + i] == acc[i]) ? 1 : 0;
      m += ok;
    }
    na = 1 + m;
  }
  #pragma unroll
  for (int j = 0; j <= Lg; ++j) out[OUT_ACCEPTED + b * (Lg + 1) + j] = (float)acc[j];
  out[OUT_NUMACC + b] = (float)na;
  row_g[b] = (b < NCg) ? gather_ids_ctx[b]
                       : (NCTX + (b - NCg) * (Lg + 1) + na - 1);
  const int last = acc[na - 1];        // == ids[gather0[b]] in the reference
  ids[b] = last;
  out[OUT_NEXTNEW + b * (Lg + 1) + 0] = (float)last;
}

// ---------------------------------------------------------------------------
// 64xNN = 64xK @ KxNN fp32 GEMM using V_WMMA_F32_16X16X4_F32.
// NN is a compile-time constant so all B loads are immediate-offset off a
// single incremented pointer pair (no per-iteration address multiplies).
// Block = 128 threads (4 waves); wave w owns M rows [16w,16w+16).
// Each wave computes 4 adjacent 16x16 N-tiles (64 columns per block).
// MODE 0: store per-K-chunk partial tile to out_parts[blockIdx.y].
// MODE 1: non-temporal B stream (lm_head > L2) + fused row-argmax over this
//         block's 64 columns -> atomicMax(keys).
// ---------------------------------------------------------------------------
template <int MODE, int NN>
__global__ __launch_bounds__(128)
void gemm64_wmma(const float* __restrict__ A, int lda,
                 const int* __restrict__ row_map,
                 const float* __restrict__ B,
                 int k_chunk,
                 float* __restrict__ out_parts,
                 unsigned long long* __restrict__ keys) {
  const int lane = threadIdx.x & 31;
  const int wv   = threadIdx.x >> 5;   // M tile 0..3
  const int half = lane >> 4;          // K sub-phase (A/B VGPR layout halves)
  const int lh   = lane & 15;          // row (A) / col (B,C) within tile
  const int mrow = wv * 16 + lh;
  const int arow = row_map ? row_map[mrow] : mrow;
  const int k0   = blockIdx.y * k_chunk;
  const int n_base = blockIdx.x * 64;

  const float* Ap  = A + (size_t)arow * lda + k0 + 2 * half;
  const float* Bp0 = B + (size_t)(k0 + 2 * half) * NN + n_base + lh;
  const float* Bp1 = Bp0 + NN;

  v8f acc0 = {}, acc1 = {}, acc2 = {}, acc3 = {};
  const int steps = k_chunk >> 2;
  #pragma unroll 2
  for (int s = 0; s < steps; ++s) {
    v2f a, b0, b1, b2, b3;
    a.x = Ap[0];
    a.y = Ap[1];
    if (MODE == 1) {
      b0.x = __builtin_nontemporal_load(Bp0 +  0);
      b0.y = __builtin_nontemporal_load(Bp1 +  0);
      b1.x = __builtin_nontemporal_load(Bp0 + 16);
      b1.y = __builtin_nontemporal_load(Bp1 + 16);
      b2.x = __builtin_nontemporal_load(Bp0 + 32);
      b2.y = __builtin_nontemporal_load(Bp1 + 32);
      b3.x = __builtin_nontemporal_load(Bp0 + 48);
      b3.y = __builtin_nontemporal_load(Bp1 + 48);
    } else {
      b0.x = Bp0[0];  b0.y = Bp1[0];
      b1.x = Bp0[16]; b1.y = Bp1[16];
      b2.x = Bp0[32]; b2.y = Bp1[32];
      b3.x = Bp0[48]; b3.y = Bp1[48];
    }
    Ap  += 4;
    Bp0 += (size_t)4 * NN;
    Bp1 += (size_t)4 * NN;
    acc0 = __builtin_amdgcn_wmma_f32_16x16x4_f32(false, a, false, b0, (short)0, acc0, false, false);
    acc1 = __builtin_amdgcn_wmma_f32_16x16x4_f32(false, a, false, b1, (short)0, acc1, false, false);
    acc2 = __builtin_amdgcn_wmma_f32_16x16x4_f32(false, a, false, b2, (short)0, acc2, false, false);
    acc3 = __builtin_amdgcn_wmma_f32_16x16x4_f32(false, a, false, b3, (short)0, acc3, false, false);
  }

  if (MODE == 0) {
    float* outP = out_parts + (size_t)blockIdx.y * ((size_t)BSZ * NN);
    #pragma unroll
    for (int j = 0; j < 8; ++j) {
      const int row = wv * 16 + j + 8 * half;           // C layout
      float* o = outP + (size_t)row * NN + n_base + lh;
      o[0]  = acc0[j];
      o[16] = acc1[j];
      o[32] = acc2[j];
      o[48] = acc3[j];
    }
  } else {
    #pragma unroll
    for (int j = 0; j < 8; ++j) {
      float v = acc0[j]; int c = n_base + lh;
      { float nv = acc1[j]; int nc = n_base + 16 + lh; if (nv > v) { v = nv; c = nc; } }
      { float nv = acc2[j]; int nc = n_base + 32 + lh; if (nv > v) { v = nv; c = nc; } }
      { float nv = acc3[j]; int nc = n_base + 48 + lh; if (nv > v) { v = nv; c = nc; } }
      // reduce across the 16 lanes of this half (cols), keep smallest col on tie
      #pragma unroll
      for (int m = 8; m >= 1; m >>= 1) {
        float ov = __shfl_xor(v, m, 32);
        int   oc = __shfl_xor(c, m, 32);
        if (ov > v || (ov == v && oc < c)) { v = ov; c = oc; }
      }
      if (lh == 0) {
        const int row = wv * 16 + j + 8 * half;
        unsigned long long key =
            ((unsigned long long)f32_ord(v) << 32) |
            (unsigned long long)(0xFFFFFFFFu - (unsigned int)c);
        atomicMax(&keys[row], key);
      }
    }
  }
}

// ---------------------------------------------------------------------------
// Sum K-split partials (fixed order -> deterministic); optional embed + tanh.
// ---------------------------------------------------------------------------
__global__ void reduce_epilogue(const float* __restrict__ parts,
                                int do_tanh_embed,
                                const int* __restrict__ ids,
                                const float* __restrict__ embed,
                                float* __restrict__ hidden) {
  const int idx = blockIdx.x * blockDim.x + threadIdx.x;
  const int TOT = BSZ * Hh;
  if (idx >= TOT) return;
  float s = parts[idx] + parts[idx + TOT] + parts[idx + 2 * TOT] + parts[idx + 3 * TOT];
  if (do_tanh_embed) {
    const int b = idx >> 11;           // / Hh
    const int j = idx & (Hh - 1);
    s = tanhf(s + embed[(size_t)ids[b] * Hh + j]);
  }
  hidden[idx] = s;
}

// ---------------------------------------------------------------------------
// Per-iteration finalize: decode argmax keys, apply d2t, write outputs.
// ---------------------------------------------------------------------------
__global__ void finalize_iter(unsigned long long* __restrict__ keys,
                              const int* __restrict__ d2t,
                              int iter,
                              float* __restrict__ out,
                              int* __restrict__ ids) {
  const int b = threadIdx.x;
  if (b >= BSZ) return;
  const unsigned long long key = keys[b];
  const int n = (int)(0xFFFFFFFFu - (unsigned int)(key & 0xFFFFFFFFu));
  const int tok = d2t[n] + n;
  out[OUT_NEXTDRAFT + b * Lg + iter] = (float)tok;
  out[OUT_NEXTNEW + b * (Lg + 1) + iter + 1] = (float)tok;
  ids[b] = tok;
  keys[b] = 0ull;   // ready for next iteration
}

// ---------------------------------------------------------------------------
extern "C" void kernel_launch(void* const* d_in, const int* in_sizes, int n_in,
                              void* d_out, int out_size, void* d_ws, size_t ws_size,
                              hipStream_t stream) {
  const float* logits          = (const float*)d_in[0];
  // d_in[1] input_ids, d_in[2] position_ids: provably unused (every gathered
  // ids position is overwritten with a target token; pos is never returned)
  const float* captured_hidden = (const float*)d_in[3];
  const int*   draft_tokens    = (const int*)d_in[4];
  const int*   gather_ids_ctx  = (const int*)d_in[5];
  const float* W_fc            = (const float*)d_in[6];
  const float* embed           = (const float*)d_in[7];
  const float* W_h             = (const float*)d_in[8];
  const float* lm_head         = (const float*)d_in[9];
  const int*   d2t             = (const int*)d_in[10];
  float* out = (float*)d_out;

  char* ws = (char*)d_ws;
  unsigned long long* keys = (unsigned long long*)(ws + 0);      // 64 * u64
  int*   tt     = (int*)(ws + 512);                              // 192 ints
  int*   row_g  = (int*)(ws + 1280);                             // 64 ints
  int*   ids    = (int*)(ws + 1536);                             // 64 ints
  float* hidden = (float*)(ws + 2048);                           // 64*2048 f32
  float* parts  = (float*)(ws + 2048 + (size_t)BSZ * Hh * 4);    // 4 * 64*2048 f32

  // 1) logits passthrough
  hipMemcpyAsync(d_out, logits, (size_t)NLROW * VTg * sizeof(float),
                 hipMemcpyDeviceToDevice, stream);

  // 2) target_tokens = argmax(logits, -1)
  argmax_rows<<<NLROW, 256, 0, stream>>>(logits, VTg, tt);

  // 3) accepted / num_accepted / gather rows / initial ids (+ zero keys)
  control_kernel<<<1, 64, 0, stream>>>(tt, draft_tokens, gather_ids_ctx,
                                       out, row_g, ids, keys);

  // 4) hidden = captured_hidden[row_g] @ W_fc   (64x6144 @ 6144x2048)
  gemm64_wmma<0, Hh><<<dim3(Hh / 64, 4), 128, 0, stream>>>(
      captured_hidden, 3 * Hh, row_g, W_fc, (3 * Hh) / 4, parts, nullptr);
  reduce_epilogue<<<(BSZ * Hh) / 256, 256, 0, stream>>>(parts, 0, nullptr,
                                                        nullptr, hidden);

  // 5) draft loop
  for (int it = 0; it < Lg; ++it) {
    // parts = hidden @ W_h (K-split)
    gemm64_wmma<0, Hh><<<dim3(Hh / 64, 4), 128, 0, stream>>>(
        hidden, Hh, nullptr, W_h, Hh / 4, parts, nullptr);
    // hidden = tanh(embed[ids] + sum(parts))
    reduce_epilogue<<<(BSZ * Hh) / 256, 256, 0, stream>>>(parts, 1, ids, embed,
                                                          hidden);
    // fused (hidden @ lm_head) + row argmax -> keys  (NT weight stream)
    gemm64_wmma<1, VDg><<<dim3(VDg / 64, 1), 128, 0, stream>>>(
        hidden, Hh, nullptr, lm_head, Hh, nullptr, keys);
    // tok = d2t[argmax] + argmax; write outputs; ids = tok
    finalize_iter<<<1, 64, 0, stream>>>(keys, d2t, it, out, ids);
  }
}